// KBLNScorer_3925600108869
// MI455X (gfx1250) — compile-verified
//
#include <hip/hip_runtime.h>
#include <math.h>

// ---------------------------------------------------------------------------
// KBLN scorer for MI455X (gfx1250, wave32).
//   out[b,e] = sum_f w[rel_b,f] * exp(-((hl[b,f]-c[f]) - lit[e,f])^2 / var[f])
// Rewritten as: p = exp2(-(lq - aq)^2),  lq = lit*q,  aq = (hl-c)*q,
//               q = sqrt(log2(e)/var)   (var > 0 by construction)
// Inner loop per 2 elements: 1 v_pk_add_f32 (neg) + 1 v_pk_mul_f32 (neg-sq)
// + 2 v_exp_f32 (TRANS, the roofline bottleneck: 204.8M exps total);
// f-reduction rides the matrix pipe via V_WMMA_F32_16X16X4_F32 with p
// computed natively in the 16x4 f32 A-matrix VGPR layout (zero shuffles).
// aq / w kept in SEPARATE LDS arrays so ds_load_b64 lands directly as the
// packed-VALU source pair and the even-aligned WMMA B operand (no v_movs).
// ---------------------------------------------------------------------------

typedef __attribute__((ext_vector_type(2))) float v2f;
typedef __attribute__((ext_vector_type(8))) float v8f;

#define NUM_E 50000
#define NUM_F 64
#define NUM_B 64
#define NTILES (NUM_E / 16)      // 3125, exact
#define WAVES_PER_BLOCK 8
#define GRID_MAIN ((NTILES + WAVES_PER_BLOCK - 1) / WAVES_PER_BLOCK)  // 391

#define LOG2E 1.4426950408889634f

// Workspace layout (floats): aq[4096] | w[4096] | q[64]
#define WS_AQ 0
#define WS_W  (NUM_B * NUM_F)
#define WS_Q  (2 * NUM_B * NUM_F)

// ---------------------------------------------------------------------------
// Kernel 1: build per-(b,f) aq / w tables and q[f] into workspace.
//   aq[b*64+f] = (lit[head[b],f] - c[f]) * q[f],  w[b*64+f] = nfw[rel[b],f]
// ---------------------------------------------------------------------------
__global__ void kbln_precompute(const float* __restrict__ lit,
                                const float* __restrict__ c,
                                const float* __restrict__ var,
                                const float* __restrict__ nfw,
                                const int* __restrict__ head_ids,
                                const int* __restrict__ rel_ids,
                                float* __restrict__ ws) {
  __shared__ float sq[NUM_F];
  int t = threadIdx.x;                      // 256 threads, 1 block
  if (t < NUM_F) {
    float q = sqrtf(LOG2E / var[t]);
    sq[t] = q;
    ws[WS_Q + t] = q;
  }
  __syncthreads();
  for (int idx = t; idx < NUM_B * NUM_F; idx += 256) {
    int b = idx >> 6;
    int f = idx & 63;
    float A = lit[(size_t)head_ids[b] * NUM_F + f] - c[f];
    ws[WS_AQ + idx] = A * sq[f];
    ws[WS_W + idx]  = nfw[(size_t)rel_ids[b] * NUM_F + f];
  }
}

// ---------------------------------------------------------------------------
// Kernel 2: main scorer. One wave per 16-e tile; loop over all 64 b.
// Lane mapping matches the 16x4 f32 WMMA A-matrix layout:
//   e = lane & 15 (row M), h = lane >> 4, A-vgpr r in {0,1} -> K = 2h + r,
//   f = 4*k_chunk + K.
// ---------------------------------------------------------------------------
__global__ void __launch_bounds__(256)
kbln_score(const float* __restrict__ lit,
           const float* __restrict__ ws,
           float* __restrict__ out) {
  __shared__ float s_aq[NUM_B * NUM_F];  // 16 KB
  __shared__ float s_w[NUM_B * NUM_F];   // 16 KB
  __shared__ float sq[NUM_F];

  int t = threadIdx.x;
  for (int i = t; i < NUM_B * NUM_F; i += 256) {
    s_aq[i] = ws[WS_AQ + i];
    s_w[i]  = ws[WS_W + i];
  }
  if (t < NUM_F) sq[t] = ws[WS_Q + t];
  __syncthreads();

  int wave = t >> 5;
  int lane = t & 31;
  int tile = blockIdx.x * WAVES_PER_BLOCK + wave;
  if (tile >= NTILES) return;            // wave-uniform: EXEC stays all-1s

  int e = tile * 16 + (lane & 15);       // matrix row M
  int h = lane >> 4;                     // 0: K=0,1  1: K=2,3
  int fo = 2 * h;                        // lane's f offset within a K-chunk

  // Stage lq = lit[e,f] * q[f] for this lane's 32 f-slots into VGPRs.
  const float* lrow = lit + (size_t)e * NUM_F;
  float lq[32];
#pragma unroll
  for (int k = 0; k < 16; ++k) {
    int f = 4 * k + fo;
    float2 lv = *reinterpret_cast<const float2*>(lrow + f);     // 8B aligned
    float2 qq = *reinterpret_cast<const float2*>(&sq[f]);
    lq[2 * k]     = lv.x * qq.x;
    lq[2 * k + 1] = lv.y * qq.y;
  }

  for (int b = 0; b < NUM_B; ++b) {
    v8f acc = {};                                  // C/D accumulator
    const float* pa = s_aq + b * NUM_F + fo;
    const float* pw = s_w  + b * NUM_F + fo;
#pragma unroll
    for (int k = 0; k < 16; ++k) {
      float2 aq = *reinterpret_cast<const float2*>(pa + 4 * k); // ds_load_b64
      v2f Bm    = *reinterpret_cast<const v2f*>(pw + 4 * k);    // ds_load_b64
      float u0 = lq[2 * k]     - aq.x;             // v_pk_add_f32 (neg)
      float u1 = lq[2 * k + 1] - aq.y;
      v2f A;
      A[0] = __builtin_amdgcn_exp2f(-(u0 * u0));   // v_pk_mul + 2x v_exp_f32
      A[1] = __builtin_amdgcn_exp2f(-(u1 * u1));
      acc = __builtin_amdgcn_wmma_f32_16x16x4_f32(
          /*neg_a=*/false, A, /*neg_b=*/false, Bm,
          /*c_mod=*/(short)0, acc, /*reuse_a=*/false, /*reuse_b=*/false);
    }
    // Column N=0 lives in lanes 0 (M=g) and 16 (M=g+8): 2x b128 stores each.
    if ((lane & 15) == 0) {
      float4* o4 = reinterpret_cast<float4*>(out + (size_t)b * NUM_E +
                                             tile * 16 + 8 * h); // 16B aligned
      o4[0] = make_float4(acc[0], acc[1], acc[2], acc[3]);
      o4[1] = make_float4(acc[4], acc[5], acc[6], acc[7]);
    }
  }
}

// ---------------------------------------------------------------------------
// Launch. Inputs (setup_inputs order):
//   0 numerical_literals f32[50000*64], 1 c f32[64], 2 var f32[64],
//   3 nf_weights f32[500*64], 4 head_ids i32[64], 5 rel_ids i32[64]
// Output: f32[64*50000]. Workspace: 33 KB (aq | w | q).
// ---------------------------------------------------------------------------
extern "C" void kernel_launch(void* const* d_in, const int* in_sizes, int n_in,
                              void* d_out, int out_size, void* d_ws, size_t ws_size,
                              hipStream_t stream) {
  const float* lit  = (const float*)d_in[0];
  const float* c    = (const float*)d_in[1];
  const float* var  = (const float*)d_in[2];
  const float* nfw  = (const float*)d_in[3];
  const int*   head = (const int*)d_in[4];
  const int*   rel  = (const int*)d_in[5];
  float* ws = (float*)d_ws;

  kbln_precompute<<<1, 256, 0, stream>>>(lit, c, var, nfw, head, rel, ws);
  kbln_score<<<GRID_MAIN, 256, 0, stream>>>(lit, ws, (float*)d_out);
}